// ParallelInputEmbedder_30090540875864
// MI455X (gfx1250) — compile-verified
//
#include <hip/hip_runtime.h>
#include <hip/hip_bf16.h>

// ---------------------------------------------------------------------------
// AlphaFold InputEmbedder for MI455X (gfx1250, wave32).
// msa GEMM runs on V_WMMA_F32_16X16X4_F32; the two big output tensors
// (403 MB total, > 192 MB L2) are written with non-temporal stores (32b for msa, 128b for pair).
// ---------------------------------------------------------------------------

typedef __attribute__((ext_vector_type(2))) float v2f;
typedef __attribute__((ext_vector_type(8))) float v8f;
typedef __attribute__((ext_vector_type(4))) float f32x4;

#define N_RES   768
#define S_SEQ   128
#define TFD     22
#define MSAD    49
#define CZ      128
#define CM      256
#define RKCLIP  32      // RELPOS_K
#define NBINS   65

#if defined(__has_builtin)
#if __has_builtin(__builtin_amdgcn_wmma_f32_16x16x4_f32)
#define HAVE_WMMA_F32X4 1
#endif
#endif

// ---------------------------------------------------------------------------
// Kernel 1: tiny tf projections  (768 x 22) @ (22 x {128,128,256})
// One thread per output element; 22 FMAs each.  Writes ei / ej / tm into ws.
// ---------------------------------------------------------------------------
__global__ void tf_proj_kernel(const float* __restrict__ tf,
                               const float* __restrict__ w_zi, const float* __restrict__ b_zi,
                               const float* __restrict__ w_zj, const float* __restrict__ b_zj,
                               const float* __restrict__ w_m,  const float* __restrict__ b_m,
                               float* __restrict__ ei, float* __restrict__ ej,
                               float* __restrict__ tm)
{
    int idx = blockIdx.x * blockDim.x + threadIdx.x;      // 0 .. 393215
    const int NEI = N_RES * CZ;                           // 98304
    const float* w; const float* b; float* dst;
    int n, c, C;
    if (idx < NEI)            { n = idx / CZ;  c = idx % CZ;  C = CZ; w = w_zi; b = b_zi; dst = ei; }
    else if (idx < 2 * NEI)   { int t = idx - NEI;
                                n = t / CZ;   c = t % CZ;   C = CZ; w = w_zj; b = b_zj; dst = ej; }
    else                      { int t = idx - 2 * NEI;
                                n = t / CM;   c = t % CM;   C = CM; w = w_m;  b = b_m;  dst = tm; }
    float acc = b[c];
    #pragma unroll
    for (int k = 0; k < TFD; ++k)
        acc = fmaf(tf[n * TFD + k], w[k * C + c], acc);
    dst[(idx < 2 * NEI) ? (n * CZ + c) : (n * CM + c)] = acc;
}

// ---------------------------------------------------------------------------
// Kernel 2: msa_emb = msa @ w_msa_m + b_msa_m + tf_m   via WMMA f32 16x16x4.
// One wave per 16-row tile of the flattened [98304, 49] A matrix.
// A is preloaded (13 K-steps x 2 VGPRs/lane), then we sweep the 16 column
// tiles of 16 channels each, accumulating 13 wmma ops per tile.
//
// Layouts per ISA 7.12.2 (wave32):
//   A 16x4 f32:  lanes 0-15 hold M=lane,    v0=K0, v1=K1
//                lanes 16-31 hold M=lane-16, v0=K2, v1=K3
//   B 4x16 f32:  lanes 0-15 hold N=lane,    v0=K0, v1=K1 (mirror of A)
//   C/D 16x16:   lane n + (m>=8 ? 16 : 0), VGPR v = m % 8
// ---------------------------------------------------------------------------
__global__ void msa_wmma_kernel(const float* __restrict__ msa,
                                const float* __restrict__ w,    // [49, 256]
                                const float* __restrict__ bias, // [256]
                                const float* __restrict__ tm,   // [768, 256]
                                float* __restrict__ out)        // [98304, 256]
{
    const int lane  = threadIdx.x & 31;
    const int rtile = blockIdx.x * (blockDim.x >> 5) + (threadIdx.x >> 5); // 0..6143
    const int m     = lane & 15;
    const int khalf = lane >> 4;                 // 0 or 1
    const long row  = (long)rtile * 16 + m;      // A row this lane supplies

    // Preload A fragments for all 13 K-steps (zero-pad K=49..51).
    float a0[13], a1[13];
    #pragma unroll
    for (int s = 0; s < 13; ++s) {
        int k0 = s * 4 + khalf * 2;
        a0[s] = (k0     < MSAD) ? msa[row * MSAD + k0]     : 0.0f;
        a1[s] = (k0 + 1 < MSAD) ? msa[row * MSAD + k0 + 1] : 0.0f;
    }

    for (int ct = 0; ct < 16; ++ct) {
        const int c = ct * 16 + m;               // output channel for B/D role
        v8f acc = {};
#if defined(HAVE_WMMA_F32X4)
        #pragma unroll
        for (int s = 0; s < 13; ++s) {
            int k0 = s * 4 + khalf * 2;
            float b0 = (k0     < MSAD) ? w[k0 * CM + c]       : 0.0f;
            float b1 = (k0 + 1 < MSAD) ? w[(k0 + 1) * CM + c] : 0.0f;
            v2f A = { a0[s], a1[s] };
            v2f B = { b0, b1 };
            // (neg_a, A, neg_b, B, c_mod, C, reuse_a, reuse_b)
            acc = __builtin_amdgcn_wmma_f32_16x16x4_f32(
                false, A, false, B, (short)0, acc, false, false);
        }
#else
        // Scalar fallback: lane owns D rows (khalf*8 + v), col c.
        #pragma unroll
        for (int v = 0; v < 8; ++v) {
            long r = (long)rtile * 16 + khalf * 8 + v;
            float sum = 0.0f;
            for (int k = 0; k < MSAD; ++k)
                sum = fmaf(msa[r * MSAD + k], w[k * CM + c], sum);
            acc[v] = sum;
        }
#endif
        // Epilogue: + bias + tf_m, non-temporal store of the 16x16 tile.
        const float bv = bias[c];
        #pragma unroll
        for (int v = 0; v < 8; ++v) {
            long r    = (long)rtile * 16 + khalf * 8 + v;  // flat (s*768 + n)
            int  nres = (int)(r % N_RES);
            float val = acc[v] + bv + tm[nres * CM + c];
            __builtin_nontemporal_store(val, out + r * CM + c);
        }
    }
}

// ---------------------------------------------------------------------------
// Kernel 3: pair_emb[i][j][c] = w_relpos[bin(i,j)][c] + b_relpos[c]
//                              + ei[i][c] + ej[j][c]
// Block = (i, 64-wide j chunk).  base[c] = ei[i][c] + b_relpos[c] in LDS.
// 302 MB of NT float4 stores; gathers stay hot in L2 (w_relpos 32KB, ej 384KB).
// ---------------------------------------------------------------------------
__global__ void pair_kernel(const int*   __restrict__ ri,
                            const float* __restrict__ w_rel,  // [65, 128]
                            const float* __restrict__ ej,     // [768, 128]
                            const float* __restrict__ ei,     // [768, 128]
                            const float* __restrict__ b_rel,  // [128]
                            float* __restrict__ out)          // [768, 768, 128]
{
    __shared__ __align__(16) float base[CZ];
    __shared__ int rii;
    const int i = blockIdx.x;
    if (threadIdx.x < CZ)
        base[threadIdx.x] = ei[i * CZ + threadIdx.x] + b_rel[threadIdx.x];
    if (threadIdx.x == 0) rii = ri[i];
    __syncthreads();

    const int jloc = threadIdx.x >> 5;          // 0..7
    const int c    = (threadIdx.x & 31) * 4;    // float4 channel base
    const f32x4 bse = *(const f32x4*)&base[c];

    #pragma unroll
    for (int jj = 0; jj < 8; ++jj) {
        const int j = blockIdx.y * 64 + jj * 8 + jloc;
        int d = rii - ri[j];
        d = (d < -RKCLIP) ? -RKCLIP : ((d > RKCLIP) ? RKCLIP : d);
        const int bin = d + RKCLIP;
        const f32x4 rp = *(const f32x4*)&w_rel[bin * CZ + c];
        const f32x4 ev = *(const f32x4*)&ej[j * CZ + c];
        f32x4 o = bse + rp + ev;
        __builtin_nontemporal_store(o, (f32x4*)(out + ((long)i * N_RES + j) * CZ + c));
    }
}

// ---------------------------------------------------------------------------
extern "C" void kernel_launch(void* const* d_in, const int* in_sizes, int n_in,
                              void* d_out, int out_size, void* d_ws, size_t ws_size,
                              hipStream_t stream) {
    const float* tf       = (const float*)d_in[0];
    const int*   ri       = (const int*)  d_in[1];
    const float* msa      = (const float*)d_in[2];
    const float* w_tf_zi  = (const float*)d_in[3];
    const float* b_tf_zi  = (const float*)d_in[4];
    const float* w_tf_zj  = (const float*)d_in[5];
    const float* b_tf_zj  = (const float*)d_in[6];
    const float* w_tf_m   = (const float*)d_in[7];
    const float* b_tf_m   = (const float*)d_in[8];
    const float* w_msa_m  = (const float*)d_in[9];
    const float* b_msa_m  = (const float*)d_in[10];
    const float* w_relpos = (const float*)d_in[11];
    const float* b_relpos = (const float*)d_in[12];

    float* ws = (float*)d_ws;
    float* ei = ws;                                // [768, 128]
    float* ej = ws + N_RES * CZ;                   // [768, 128]
    float* tm = ws + 2 * N_RES * CZ;               // [768, 256]

    float* out_msa  = (float*)d_out;                           // [128,768,256]
    float* out_pair = out_msa + (long)S_SEQ * N_RES * CM;      // [768,768,128]

    // 1) tiny tf projections -> workspace  (393216 threads)
    tf_proj_kernel<<<1536, 256, 0, stream>>>(tf, w_tf_zi, b_tf_zi, w_tf_zj, b_tf_zj,
                                             w_tf_m, b_tf_m, ei, ej, tm);

    // 2) msa GEMM + fused epilogue via WMMA: 6144 row tiles, 4 waves/block
    msa_wmma_kernel<<<1536, 128, 0, stream>>>(msa, w_msa_m, b_msa_m, tm, out_msa);

    // 3) pair broadcast-add: grid (i, j-chunk of 64)
    pair_kernel<<<dim3(N_RES, N_RES / 64), 256, 0, stream>>>(ri, w_relpos, ej, ei,
                                                             b_relpos, out_pair);
}